// Decoder_26104811225843
// MI455X (gfx1250) — compile-verified
//
#include <hip/hip_runtime.h>
#include <hip/hip_bf16.h>
#include <math.h>

typedef float v2f __attribute__((ext_vector_type(2)));
typedef float v8f __attribute__((ext_vector_type(8)));

#define N_NODES 50000
#define N_EDGES 800000
#define D_IN    32
#define D_OUT   64

// ---------------------------------------------------------------------------
// Kernel 1: xw = x @ W   via fp32 WMMA (V_WMMA_F32_16X16X4_F32)
// One wave (32 lanes) computes one 16x16 output tile; K=32 in 8 steps of 4.
// ---------------------------------------------------------------------------
__global__ __launch_bounds__(32) void gemm_xw_wmma(const float* __restrict__ x,
                                                   const float* __restrict__ W,
                                                   float* __restrict__ xw) {
    const int lane  = threadIdx.x & 31;
    const int hi    = lane >> 4;      // 0: lanes 0-15, 1: lanes 16-31
    const int l     = lane & 15;
    const int mbase = blockIdx.x * 16;   // 3125 tiles, exact
    const int nbase = blockIdx.y * 16;   // 4 tiles

    const int row = mbase + l;           // A-matrix row for this lane
    const int col = nbase + l;           // B-matrix column for this lane

    v8f c = {};
#pragma unroll
    for (int k0 = 0; k0 < D_IN; k0 += 4) {
        const int kb = k0 + hi * 2;      // lanes 16-31 carry K+2..K+3
        v2f a, bm;
        // A 16x4 layout: VGPR0 = K=kb, VGPR1 = K=kb+1 (per half-wave)
        a.x  = x[row * D_IN + kb];
        a.y  = x[row * D_IN + kb + 1];
        // B 4x16 layout: VGPR v = row K=v (lanes 0-15) / K=v+2 (lanes 16-31)
        bm.x = W[(kb)     * D_OUT + col];
        bm.y = W[(kb + 1) * D_OUT + col];
        c = __builtin_amdgcn_wmma_f32_16x16x4_f32(
                /*neg_a=*/false, a, /*neg_b=*/false, bm,
                /*c_mod=*/(short)0, c, /*reuse_a=*/false, /*reuse_b=*/false);
    }

    // C/D layout: VGPR j holds M = j (lanes 0-15) / M = j+8 (lanes 16-31)
#pragma unroll
    for (int j = 0; j < 8; ++j) {
        xw[(size_t)(mbase + j + hi * 8) * D_OUT + nbase + l] = c[j];
    }
}

// ---------------------------------------------------------------------------
// Kernel 2: deg[i] = 1.0 (self-loop contribution)
// ---------------------------------------------------------------------------
__global__ void deg_init(float* __restrict__ deg) {
    int i = blockIdx.x * blockDim.x + threadIdx.x;
    if (i < N_NODES) deg[i] = 1.0f;
}

// ---------------------------------------------------------------------------
// Kernel 3: deg[dst[e]] += 1 over all edges
// ---------------------------------------------------------------------------
__global__ void deg_edges(const int* __restrict__ dst, float* __restrict__ deg) {
    int e = blockIdx.x * blockDim.x + threadIdx.x;
    if (e < N_EDGES) atomicAdd(&deg[dst[e]], 1.0f);
}

// ---------------------------------------------------------------------------
// Kernel 4: deg[i] <- rsqrt(deg[i])   (deg >= 1 always, matches dinv)
// ---------------------------------------------------------------------------
__global__ void deg_to_dinv(float* __restrict__ deg) {
    int i = blockIdx.x * blockDim.x + threadIdx.x;
    if (i < N_NODES) {
        float d = deg[i];
        deg[i] = (d > 0.0f) ? rsqrtf(d) : 0.0f;
    }
}

// ---------------------------------------------------------------------------
// Kernel 5: h init with self-loop message: h[i,:] = xw[i,:] * dinv[i]^2
// ---------------------------------------------------------------------------
__global__ void h_init_selfloop(const float* __restrict__ xw,
                                const float* __restrict__ dinv,
                                float* __restrict__ h) {
    int i = blockIdx.x * blockDim.x + threadIdx.x;
    if (i < N_NODES * D_OUT) {
        int node = i >> 6;  // D_OUT == 64
        float s = dinv[node];
        h[i] = xw[i] * s * s;
    }
}

// ---------------------------------------------------------------------------
// Kernel 6: edge scatter-add: h[dst,:] += xw[src,:] * dinv[src]*dinv[dst]
// One wave per edge; each lane handles 2 of the 64 feature dims (coalesced).
// ---------------------------------------------------------------------------
__global__ __launch_bounds__(256) void scatter_msgs(const int* __restrict__ src,
                                                    const int* __restrict__ dst,
                                                    const float* __restrict__ xw,
                                                    const float* __restrict__ dinv,
                                                    float* __restrict__ h) {
    int e    = blockIdx.x * 8 + (threadIdx.x >> 5);   // 8 waves per block
    int lane = threadIdx.x & 31;
    if (e >= N_EDGES) return;
    int s = src[e];
    int d = dst[e];
    float norm = dinv[s] * dinv[d];
    const float2 v = *(const float2*)&xw[(size_t)s * D_OUT + lane * 2];
    float* hp = &h[(size_t)d * D_OUT + lane * 2];
    atomicAdd(hp + 0, v.x * norm);
    atomicAdd(hp + 1, v.y * norm);
}

// ---------------------------------------------------------------------------
// Kernel 7: h = relu(h + b)
// ---------------------------------------------------------------------------
__global__ void bias_relu(float* __restrict__ h, const float* __restrict__ b) {
    int i = blockIdx.x * blockDim.x + threadIdx.x;
    if (i < N_NODES * D_OUT) {
        float v = h[i] + b[i & (D_OUT - 1)];
        h[i] = v > 0.0f ? v : 0.0f;
    }
}

// ---------------------------------------------------------------------------
// Kernel 8: decoder: adj[e] = sigmoid(<h[src], h[dst]>) + 1e-15
// One wave per edge; lane covers 2 dims; wave32 shuffle reduction.
// ---------------------------------------------------------------------------
__global__ __launch_bounds__(256) void decoder(const int* __restrict__ src,
                                               const int* __restrict__ dst,
                                               const float* __restrict__ h,
                                               float* __restrict__ adj) {
    int e    = blockIdx.x * 8 + (threadIdx.x >> 5);
    int lane = threadIdx.x & 31;
    if (e >= N_EDGES) return;
    int s = src[e];
    int d = dst[e];
    const float2 ps = *(const float2*)&h[(size_t)s * D_OUT + lane * 2];
    const float2 pd = *(const float2*)&h[(size_t)d * D_OUT + lane * 2];
    float partial = ps.x * pd.x + ps.y * pd.y;
#pragma unroll
    for (int m = 16; m >= 1; m >>= 1)
        partial += __shfl_xor(partial, m, 32);
    if (lane == 0)
        adj[e] = 1.0f / (1.0f + __expf(-partial)) + 1e-15f;
}

// ---------------------------------------------------------------------------
// Kernel 9: pass edge_index through to the second output slot
// ---------------------------------------------------------------------------
__global__ void copy_edges(const int* __restrict__ ei, int* __restrict__ out) {
    int i = blockIdx.x * blockDim.x + threadIdx.x;
    if (i < 2 * N_EDGES) out[i] = ei[i];
}

// ---------------------------------------------------------------------------
extern "C" void kernel_launch(void* const* d_in, const int* in_sizes, int n_in,
                              void* d_out, int out_size, void* d_ws, size_t ws_size,
                              hipStream_t stream) {
    const float* x  = (const float*)d_in[0];   // [N, 32]
    const int*   ei = (const int*)  d_in[1];   // [2, E]
    const float* W  = (const float*)d_in[2];   // [32, 64]
    const float* b  = (const float*)d_in[3];   // [64]

    const int* src = ei;             // edge_index[0]
    const int* dst = ei + N_EDGES;   // edge_index[1]

    char* ws = (char*)d_ws;
    float* xw  = (float*)ws; ws += (size_t)N_NODES * D_OUT * sizeof(float);
    float* h   = (float*)ws; ws += (size_t)N_NODES * D_OUT * sizeof(float);
    float* deg = (float*)ws; ws += (size_t)N_NODES * sizeof(float);  // reused as dinv

    float* adj    = (float*)d_out;          // first E floats
    int*   out_ei = ((int*)d_out) + N_EDGES; // next 2E ints

    // 1. xw = x @ W  (WMMA fp32): 3125 M-tiles x 4 N-tiles, 1 wave each
    gemm_xw_wmma<<<dim3(N_NODES / 16, D_OUT / 16), 32, 0, stream>>>(x, W, xw);

    // 2-4. symmetric normalization coefficients
    deg_init   <<<(N_NODES + 255) / 256, 256, 0, stream>>>(deg);
    deg_edges  <<<(N_EDGES + 255) / 256, 256, 0, stream>>>(dst, deg);
    deg_to_dinv<<<(N_NODES + 255) / 256, 256, 0, stream>>>(deg);

    // 5. self-loop messages initialize h
    h_init_selfloop<<<(N_NODES * D_OUT + 255) / 256, 256, 0, stream>>>(xw, deg, h);

    // 6. edge messages scatter-add into h
    scatter_msgs<<<(N_EDGES + 7) / 8, 256, 0, stream>>>(src, dst, xw, deg, h);

    // 7. bias + relu
    bias_relu<<<(N_NODES * D_OUT + 255) / 256, 256, 0, stream>>>(h, b);

    // 8. inner-product decoder
    decoder<<<(N_EDGES + 7) / 8, 256, 0, stream>>>(src, dst, h, adj);

    // 9. edge_index passthrough output
    copy_edges<<<(2 * N_EDGES + 255) / 256, 256, 0, stream>>>(ei, out_ei);
}